// WKV_40106404610731
// MI455X (gfx1250) — compile-verified
//
#include <hip/hip_runtime.h>
#include <stdint.h>

// RWKV-4 WKV forward for MI455X (gfx1250).
// Bandwidth-bound streaming scan: 384 MB of traffic -> ~16.5 us floor @ 23.3 TB/s.
// Per-channel scalar recurrence => no WMMA structure; the CDNA5 feature that
// pays here is the Tensor Data Mover: DMA [TCHUNK x CT] strided k/v tiles into
// LDS with a 3-deep pipeline (TENSORcnt), keeping ~8 MB of HBM traffic in
// flight across 256 blocks while the vector pipe runs the exp/max/fma scan.

typedef uint32_t u32;
typedef uint64_t u64;
typedef __attribute__((ext_vector_type(4))) unsigned int uv4;
typedef __attribute__((ext_vector_type(8))) int         iv8;
typedef __attribute__((ext_vector_type(4))) int         iv4;

#define BDIM   16
#define TDIM   1024
#define CDIM   2048
#define CT     128              // channels per block (one thread per channel)
#define TCHUNK 16               // timesteps per LDS tile
#define NBUF   3                // TDM pipeline depth (2 chunks prefetch ahead)
#define NCHUNK (TDIM / TCHUNK)  // 64 chunks

// Issue one TDM load of a [TCHUNK x CT] float tile (row stride CDIM floats)
// from global memory into LDS at byte offset lds_off.
__device__ __forceinline__ void tdm_load_tile(u32 lds_off, const float* gptr) {
    u64 ga = (u64)(uintptr_t)gptr;

    // D# group 0: count=1 user descriptor, lds_addr, global_addr[56:0], type=2.
    uv4 g0;
    g0.x = 1u;                                   // count=1, is_restore=0, gather off
    g0.y = lds_off;                              // LDS byte address
    g0.z = (u32)(ga & 0xFFFFFFFFu);              // global_addr[31:0]
    g0.w = (u32)((ga >> 32) & 0x01FFFFFFu)       // global_addr[56:32]
         | (2u << 30);                           // type = 2 ("image")

    // D# group 1: data_size=4B, tensor dims, tile dims, dim0 stride.
    iv8 g1;
    g1[0] = (int)(2u << 16);                     // data_size = 2 -> 4 bytes; wg_mask=0
    g1[1] = (int)((u32)CDIM << 16);              // tensor_dim0[15:0]  = 2048 (bits 79:48)
    g1[2] = (int)((u32)(BDIM * TDIM) << 16);     // tensor_dim1[15:0]  = 16384 (bits 111:80)
    g1[3] = (int)((u32)CT << 16);                // tile_dim0 = 128 (bits 127:112)
    g1[4] = (int)TCHUNK;                         // tile_dim1 = 16 (bits 143:128)
    g1[5] = (int)CDIM;                           // tensor_dim0_stride[31:0] = 2048 elems
    g1[6] = 0;                                   // stride hi / dim1_stride lo
    g1[7] = 0;

    iv4 gz4 = {0, 0, 0, 0};                      // groups 2/3 unused (2D tensor)
    iv8 gz8 = {0, 0, 0, 0, 0, 0, 0, 0};          // extra group (clang-23 6-arg form)

    __builtin_amdgcn_tensor_load_to_lds(g0, g1, gz4, gz4, gz8, 0);
}

__global__ __launch_bounds__(CT) void WKV_40106404610731_kernel(
    const float* __restrict__ w, const float* __restrict__ u,
    const float* __restrict__ k, const float* __restrict__ v,
    float* __restrict__ y)
{
    // [buffer][k/v][TCHUNK*CT] : 3*2*16*128*4B = 48 KB
    __shared__ float smem[NBUF][2][TCHUNK * CT];

    const int tid = threadIdx.x;
    const int b   = blockIdx.x;          // 0..15
    const int c0  = blockIdx.y * CT;     // channel tile base
    const int c   = c0 + tid;

    const float wc = w[c];
    const float uc = u[c];

    // Recurrence state lives in registers for the whole T loop.
    float p = 0.0f, q = 0.0f, o = -1e38f;

    const size_t tile_base = ((size_t)b * TDIM) * CDIM + (size_t)c0; // (b, t=0, c0)
    const float* kp = k + tile_base;
    const float* vp = v + tile_base;
    float* yp = y + ((size_t)b * TDIM) * CDIM + (size_t)c;

    const u32 lds_base   = (u32)(uintptr_t)(&smem[0][0][0]);
    const u32 tile_bytes = (u32)(TCHUNK * CT * sizeof(float));

    const bool issuer = (tid < 32);      // wave 0 owns the TDM (EXEC ignored by TDM)

    // Prologue: start chunks 0..NBUF-2 into buffers 0..NBUF-2.
    if (issuer) {
        #pragma unroll
        for (int d = 0; d < NBUF - 1; ++d) {
            const u32 off = lds_base + (u32)d * 2u * tile_bytes;
            tdm_load_tile(off,              kp + (size_t)d * TCHUNK * CDIM);
            tdm_load_tile(off + tile_bytes, vp + (size_t)d * TCHUNK * CDIM);
        }
    }

    for (int ch = 0; ch < NCHUNK; ++ch) {
        const int cur = ch % NBUF;

        if (issuer) {
            const int ahead = ch + (NBUF - 1);
            if (ahead < NCHUNK) {
                // Buffer (ahead % NBUF) == buffer consumed at iteration ch-1;
                // the trailing barrier of ch-1 guarantees it is free.
                const u32 off = lds_base + (u32)(ahead % NBUF) * 2u * tile_bytes;
                tdm_load_tile(off,              kp + (size_t)ahead * TCHUNK * CDIM);
                tdm_load_tile(off + tile_bytes, vp + (size_t)ahead * TCHUNK * CDIM);
                // 6 outstanding (3 chunks x 2); <=4 means chunk `ch` landed.
                __builtin_amdgcn_s_wait_tensorcnt(4);
            } else if (ch == NCHUNK - 2) {
                __builtin_amdgcn_s_wait_tensorcnt(2);   // chunks N-2,N-1 in flight
            } else {
                __builtin_amdgcn_s_wait_tensorcnt(0);   // last chunk
            }
        }
        __syncthreads();                 // chunk `ch` visible to all waves

        const float* ks = &smem[cur][0][0];
        const float* vs = &smem[cur][1][0];

        #pragma unroll
        for (int i = 0; i < TCHUNK; ++i) {
            const float kt = ks[i * CT + tid];
            const float vt = vs[i * CT + tid];

            // Output at time t (bonus u on current token), log-sum-exp stabilized.
            const float no  = fmaxf(o, uc + kt);
            const float A   = __expf(o - no);
            const float E   = __expf(uc + kt - no);
            const float yy  = (A * p + E * vt) / (A * q + E);

            // State update (decay w on history).
            const float no2 = fmaxf(wc + o, kt);
            const float A2  = __expf(wc + o - no2);
            const float E2  = __expf(kt - no2);
            p = A2 * p + E2 * vt;
            q = A2 * q + E2;
            o = no2;

            // y is written once, never re-read: non-temporal, coalesced per lane row.
            __builtin_nontemporal_store(yy, &yp[(size_t)(ch * TCHUNK + i) * CDIM]);
        }
        __syncthreads();                 // all reads of buf `cur` done before reuse
    }
}

extern "C" void kernel_launch(void* const* d_in, const int* in_sizes, int n_in,
                              void* d_out, int out_size, void* d_ws, size_t ws_size,
                              hipStream_t stream) {
    // setup_inputs() order: B, T, C (scalars), w, u, k, v
    const float* w = (const float*)d_in[3];
    const float* u = (const float*)d_in[4];
    const float* k = (const float*)d_in[5];
    const float* v = (const float*)d_in[6];
    float* y = (float*)d_out;

    dim3 grid(BDIM, CDIM / CT);          // 16 x 16 = 256 blocks, 4 waves each
    WKV_40106404610731_kernel<<<grid, CT, 0, stream>>>(w, u, k, v, y);
}